// ATTConv_27616639713356
// MI455X (gfx1250) — compile-verified
//
#include <hip/hip_runtime.h>
#include <hip/hip_bf16.h>
#include <math.h>

typedef __attribute__((ext_vector_type(2))) float v2f;
typedef __attribute__((ext_vector_type(8))) float v8f;

#define DIM 128
#define LN_EPS 1e-5f
#define SCALE 0.08838834764831844f   /* 128^-0.5 */

__device__ __forceinline__ float waveSum(float v) {
#pragma unroll
  for (int off = 16; off > 0; off >>= 1) v += __shfl_xor(v, off, 32);
  return v;
}

// Standard float atomic-max via ordered integer bit patterns.
__device__ __forceinline__ void atomicMaxF(float* addr, float v) {
  if (v >= 0.0f) atomicMax((int*)addr, __float_as_int(v));
  else           atomicMin((unsigned int*)addr, __float_as_uint(v));
}

// ---------------------------------------------------------------- init
__global__ __launch_bounds__(256) void k_init(float* __restrict__ z,
                                              float* __restrict__ m,
                                              float* __restrict__ denom,
                                              int N) {
  long long i = (long long)blockIdx.x * blockDim.x + threadIdx.x;
  long long total = (long long)N * DIM;
  if (i < total) z[i] = 0.0f;
  if (i < N) { m[i] = -INFINITY; denom[i] = 0.0f; }
}

// ------------------------------------------- fused 3-way GEMM via WMMA f32
// wh = x @ Wfc^T, tx = x @ Wth^T, px = x @ Wph^T   (all [N,128])
// Block = 256 threads = 8 waves; wave w owns output columns [16w,16w+16);
// block b owns rows [16b,16b+16). A-tile registers shared across 3 B's.
__global__ __launch_bounds__(256) void k_gemm3_wmma(
    const float* __restrict__ x,
    const float* __restrict__ Wfc, const float* __restrict__ Wth,
    const float* __restrict__ Wph,
    float* __restrict__ wh, float* __restrict__ tx, float* __restrict__ px,
    int N) {
  const int lane = threadIdx.x & 31;
  const int wave = threadIdx.x >> 5;   // 0..7 -> column tile
  const int m0 = blockIdx.x * 16;
  const int half = lane >> 4;          // 0 or 1: selects K pair
  const int lrow = lane & 15;

  // A layout (16x4 f32): lanes 0-15 hold K={k,k+1}, lanes 16-31 K={k+2,k+3},
  // row = lane&15. B (4x16) mirrors it with N = lane&15 over W's rows.
  int arow = m0 + lrow;
  if (arow >= N) arow = N - 1;         // clamp loads for tail tile (EXEC must stay all-1)
  const float* ap = x + (size_t)arow * DIM + 2 * half;
  const size_t woff = (size_t)(wave * 16 + lrow) * DIM + 2 * half;
  const float* b0p = Wfc + woff;
  const float* b1p = Wth + woff;
  const float* b2p = Wph + woff;

  v8f c0 = {}, c1 = {}, c2 = {};
#pragma unroll
  for (int k = 0; k < DIM; k += 4) {
    v2f a  = *(const v2f*)(ap  + k);
    v2f b0 = *(const v2f*)(b0p + k);
    v2f b1 = *(const v2f*)(b1p + k);
    v2f b2 = *(const v2f*)(b2p + k);
    c0 = __builtin_amdgcn_wmma_f32_16x16x4_f32(false, a, false, b0, (short)0, c0, false, false);
    c1 = __builtin_amdgcn_wmma_f32_16x16x4_f32(false, a, false, b1, (short)0, c1, false, false);
    c2 = __builtin_amdgcn_wmma_f32_16x16x4_f32(false, a, false, b2, (short)0, c2, false, false);
  }

  // C/D layout: VGPR r, lanes 0-15 -> (M=r, N=lane); lanes 16-31 -> (M=r+8, N=lane-16)
  const int scol = wave * 16 + lrow;
  const int srow0 = m0 + 8 * half;
#pragma unroll
  for (int r = 0; r < 8; ++r) {
    int srow = srow0 + r;
    if (srow < N) {
      size_t o = (size_t)srow * DIM + scol;
      wh[o] = c0[r];
      tx[o] = c1[r];
      px[o] = c2[r];
    }
  }
}

// ----------------------------------- edge scores + segment max (wave/edge)
__global__ __launch_bounds__(256) void k_edge_score(
    const float* __restrict__ tx, const float* __restrict__ px,
    const int* __restrict__ src, const int* __restrict__ dst,
    float* __restrict__ score, float* __restrict__ m, int E) {
  int e = blockIdx.x * 8 + (threadIdx.x >> 5);
  if (e >= E) return;
  int lane = threadIdx.x & 31;
  int s = src[e], d = dst[e];
  float4 t = ((const float4*)(tx + (size_t)s * DIM))[lane];
  float4 p = ((const float4*)(px + (size_t)d * DIM))[lane];
  float v = waveSum(t.x * p.x + t.y * p.y + t.z * p.z + t.w * p.w);
  if (lane == 0) {
    float a = v * SCALE;
    score[e] = a;
    atomicMaxF(m + d, a);
  }
}

// ------------------------------------------- exp(a - m[dst]) + segment sum
__global__ __launch_bounds__(256) void k_edge_exp(
    float* __restrict__ score, const int* __restrict__ dst,
    const float* __restrict__ m, float* __restrict__ denom, int E) {
  int e = blockIdx.x * blockDim.x + threadIdx.x;
  if (e >= E) return;
  int d = dst[e];
  float ev = __expf(score[e] - m[d]);   // m[d] finite: max includes this edge
  score[e] = ev;
  atomicAdd(denom + d, ev);
}

// ---------------------------- alpha * wh[src] scatter-add into z (wave/edge)
__global__ __launch_bounds__(256) void k_edge_scatter(
    const float* __restrict__ wh, const float* __restrict__ escore,
    const float* __restrict__ denom, const int* __restrict__ src,
    const int* __restrict__ dst, float* __restrict__ z, int E) {
  int e = blockIdx.x * 8 + (threadIdx.x >> 5);
  if (e >= E) return;
  int lane = threadIdx.x & 31;
  int s = src[e], d = dst[e];
  float dn = denom[d];
  float alpha = escore[e] / (dn == 0.0f ? 1.0f : dn);
  float4 w = ((const float4*)(wh + (size_t)s * DIM))[lane];
  float* zp = z + (size_t)d * DIM + lane * 4;
  atomicAdd(zp + 0, alpha * w.x);
  atomicAdd(zp + 1, alpha * w.y);
  atomicAdd(zp + 2, alpha * w.z);
  atomicAdd(zp + 3, alpha * w.w);
}

// ------------------------------------------------ LayerNorm (wave per node)
__global__ __launch_bounds__(256) void k_layernorm(
    float* __restrict__ z, const float* __restrict__ gamma,
    const float* __restrict__ beta, int N) {
  int n = blockIdx.x * 8 + (threadIdx.x >> 5);
  if (n >= N) return;
  int lane = threadIdx.x & 31;
  float4 v = ((const float4*)(z + (size_t)n * DIM))[lane];
  float mu = waveSum(v.x + v.y + v.z + v.w) * (1.0f / DIM);
  float d0 = v.x - mu, d1 = v.y - mu, d2 = v.z - mu, d3 = v.w - mu;
  float var = waveSum(d0 * d0 + d1 * d1 + d2 * d2 + d3 * d3) * (1.0f / DIM);
  float rstd = rsqrtf(var + LN_EPS);
  float4 g = ((const float4*)gamma)[lane];
  float4 b = ((const float4*)beta)[lane];
  float4 o;
  o.x = d0 * rstd * g.x + b.x;
  o.y = d1 * rstd * g.y + b.y;
  o.z = d2 * rstd * g.z + b.z;
  o.w = d3 * rstd * g.w + b.w;
  ((float4*)(z + (size_t)n * DIM))[lane] = o;
}

extern "C" void kernel_launch(void* const* d_in, const int* in_sizes, int n_in,
                              void* d_out, int out_size, void* d_ws, size_t ws_size,
                              hipStream_t stream) {
  const float* x     = (const float*)d_in[0];
  const float* Wfc   = (const float*)d_in[1];
  const float* Wth   = (const float*)d_in[2];
  const float* Wph   = (const float*)d_in[3];
  const float* gamma = (const float*)d_in[4];
  const float* beta  = (const float*)d_in[5];
  const int*   src   = (const int*)d_in[6];
  const int*   dst   = (const int*)d_in[7];

  const int N = in_sizes[0] / DIM;
  const int E = in_sizes[6];
  float* z = (float*)d_out;   // accumulate z in d_out, LN in-place

  // Workspace layout (≈161 MB): wh | tx | px | edge score/exp | m | denom
  float* wh    = (float*)d_ws;
  float* tx    = wh + (size_t)N * DIM;
  float* px    = tx + (size_t)N * DIM;
  float* es    = px + (size_t)N * DIM;
  float* m     = es + (size_t)E;
  float* denom = m + (size_t)N;

  long long initElems = (long long)N * DIM;
  int initBlocks = (int)((initElems + 255) / 256);
  k_init<<<initBlocks, 256, 0, stream>>>(z, m, denom, N);

  k_gemm3_wmma<<<(N + 15) / 16, 256, 0, stream>>>(x, Wfc, Wth, Wph, wh, tx, px, N);

  int edgeWaveBlocks = (E + 7) / 8;
  k_edge_score<<<edgeWaveBlocks, 256, 0, stream>>>(tx, px, src, dst, es, m, E);
  k_edge_exp<<<(E + 255) / 256, 256, 0, stream>>>(es, dst, m, denom, E);
  k_edge_scatter<<<edgeWaveBlocks, 256, 0, stream>>>(wh, es, denom, src, dst, z, E);

  k_layernorm<<<(N + 7) / 8, 256, 0, stream>>>(z, gamma, beta, N);
}